// GCNEncoder_5566277616086
// MI455X (gfx1250) — compile-verified
//
#include <hip/hip_runtime.h>

typedef float v2f __attribute__((ext_vector_type(2)));
typedef float v8f __attribute__((ext_vector_type(8)));

// ---------------------------------------------------------------- utilities
__global__ void zero_i32_kernel(int* __restrict__ p, int n) {
    int i = blockIdx.x * blockDim.x + threadIdx.x;
    if (i < n) p[i] = 0;
}

__global__ void count_deg_kernel(const int* __restrict__ dst, int* __restrict__ deg, int E) {
    int e = blockIdx.x * blockDim.x + threadIdx.x;
    if (e < E) atomicAdd(&deg[dst[e]], 1);
}

__global__ void dinv_kernel(const int* __restrict__ deg, float* __restrict__ dinv, int n) {
    int i = blockIdx.x * blockDim.x + threadIdx.x;
    if (i < n) dinv[i] = rsqrtf((float)deg[i] + 1.0f);
}

// Single-block exclusive prefix scan of deg[0..n) -> rowstart[0..n]
// (rowstart[n] = total).  LDS Hillis-Steele per 256-chunk with running carry.
__global__ void scan_kernel(const int* __restrict__ deg, int* __restrict__ rowstart, int n) {
    __shared__ int smem[256];
    __shared__ int carry;
    const int tid = threadIdx.x;
    if (tid == 0) carry = 0;
    __syncthreads();
    for (int base = 0; base < n; base += 256) {
        const int i = base + tid;
        const int v = (i < n) ? deg[i] : 0;
        smem[tid] = v;
        __syncthreads();
        for (int ofs = 1; ofs < 256; ofs <<= 1) {
            int t = (tid >= ofs) ? smem[tid - ofs] : 0;
            __syncthreads();
            smem[tid] += t;
            __syncthreads();
        }
        const int incl = smem[tid];
        if (i < n) rowstart[i] = carry + (incl - v);
        __syncthreads();                 // everyone has read carry
        if (tid == 0) carry += smem[255];
        __syncthreads();                 // carry updated before next chunk
    }
    if (tid == 0) rowstart[n] = carry;
}

__global__ void copy_i32_kernel(const int* __restrict__ a, int* __restrict__ b, int n) {
    int i = blockIdx.x * blockDim.x + threadIdx.x;
    if (i < n) b[i] = a[i];
}

// Bucket-fill CSR: csr_src[pos] = src, csr_norm[pos] = dinv[src]*dinv[dst]
__global__ void fill_csr_kernel(const int* __restrict__ src, const int* __restrict__ dst,
                                const float* __restrict__ dinv, int* __restrict__ cursor,
                                int* __restrict__ csr_src, float* __restrict__ csr_norm, int E) {
    int e = blockIdx.x * blockDim.x + threadIdx.x;
    if (e >= E) return;
    const int s = src[e];
    const int d = dst[e];
    const int pos = atomicAdd(&cursor[d], 1);
    csr_src[pos]  = s;
    csr_norm[pos] = dinv[s] * dinv[d];
}

// ------------------------------------------------ fp32 WMMA GEMM, row-major
// C[M,N] = A[M,K] * B[K,N].  Requires M%16==0, N%16==0, K%8==0.
// One wave32 per 16x16 C tile; two accumulators for WMMA ILP.
__global__ void gemm_f32_wmma_kernel(const float* __restrict__ A,
                                     const float* __restrict__ B,
                                     float* __restrict__ C,
                                     int M, int N, int K) {
    const int lane = threadIdx.x & 31;
    const int wave = (blockIdx.x * blockDim.x + threadIdx.x) >> 5;
    const int nTiles = N >> 4;
    const int mTiles = M >> 4;
    if (wave >= mTiles * nTiles) return;           // uniform per wave: EXEC stays full
    const int mt  = wave / nTiles;
    const int nt  = wave - mt * nTiles;
    const int l16 = lane & 15;
    const int hi  = lane >> 4;                     // 0 or 1 (half-wave)
    const int row = (mt << 4) + l16;               // A row for this lane
    const int col = (nt << 4) + l16;               // B/C column for this lane
    const int kOff = hi << 1;                      // K = 0..1 (lo half) / 2..3 (hi half)

    v8f acc0 = {};
    v8f acc1 = {};
    const float* ap = A + (size_t)row * K + kOff;
    const float* bp = B + (size_t)kOff * N + col;
    const size_t bStep4 = (size_t)4 * N;

    for (int k = 0; k < K; k += 8) {
        v2f a0 = *(const v2f*)ap;                  // A[row][k+kOff .. k+kOff+1]
        v2f b0; b0.x = bp[0]; b0.y = bp[N];        // B[k+kOff][col], B[k+kOff+1][col]
        acc0 = __builtin_amdgcn_wmma_f32_16x16x4_f32(false, a0, false, b0,
                                                     (short)0, acc0, false, false);
        v2f a1 = *(const v2f*)(ap + 4);
        v2f b1; b1.x = bp[bStep4]; b1.y = bp[bStep4 + N];
        acc1 = __builtin_amdgcn_wmma_f32_16x16x4_f32(false, a1, false, b1,
                                                     (short)0, acc1, false, false);
        ap += 8;
        bp += 2 * bStep4;
    }
    v8f acc = acc0 + acc1;

    // C/D layout: VGPR v -> M = 16*mt + v + 8*hi, N = 16*nt + (lane&15)
    const int mBase = (mt << 4) + (hi << 3);
    #pragma unroll
    for (int v = 0; v < 8; ++v) {
        C[(size_t)(mBase + v) * N + col] = acc[v];
    }
}

// ------------------------- CSR gather aggregation, F=256, one wave per node
// out[i] = act( bias + dinv[i]^2 * h[i] + sum_j norm_j * h[csr_src[j]] )
__global__ void gather256_kernel(const int* __restrict__ rowstart,
                                 const int* __restrict__ csr_src,
                                 const float* __restrict__ csr_norm,
                                 const float* __restrict__ dinv,
                                 const float* __restrict__ h,
                                 const float* __restrict__ bias,
                                 float* __restrict__ out, int n, int doRelu) {
    const int lane = threadIdx.x & 31;
    int node = (blockIdx.x * blockDim.x + threadIdx.x) >> 5;
    if (node >= n) return;
    node = __builtin_amdgcn_readfirstlane(node);   // force scalar neighbor walk

    const int beg = rowstart[node];
    const int end = rowstart[node + 1];
    const float di = dinv[node];
    const float s  = di * di;

    const int c0 = lane;          // float4 slot 0..31
    const int c1 = lane + 32;     // float4 slot 32..63
    const float4* hn = (const float4*)(h + (size_t)node * 256);
    const float4* b4 = (const float4*)bias;

    float4 hv0 = hn[c0], hv1 = hn[c1];
    float4 bv0 = b4[c0], bv1 = b4[c1];
    float4 a0, a1;
    a0.x = fmaf(s, hv0.x, bv0.x); a0.y = fmaf(s, hv0.y, bv0.y);
    a0.z = fmaf(s, hv0.z, bv0.z); a0.w = fmaf(s, hv0.w, bv0.w);
    a1.x = fmaf(s, hv1.x, bv1.x); a1.y = fmaf(s, hv1.y, bv1.y);
    a1.z = fmaf(s, hv1.z, bv1.z); a1.w = fmaf(s, hv1.w, bv1.w);

    for (int j = beg; j < end; ++j) {
        const int   sj = csr_src[j];               // wave-uniform -> scalar load
        const float nj = csr_norm[j];
        const float4* hs = (const float4*)(h + (size_t)sj * 256);
        float4 v0 = hs[c0];
        float4 v1 = hs[c1];
        a0.x = fmaf(nj, v0.x, a0.x); a0.y = fmaf(nj, v0.y, a0.y);
        a0.z = fmaf(nj, v0.z, a0.z); a0.w = fmaf(nj, v0.w, a0.w);
        a1.x = fmaf(nj, v1.x, a1.x); a1.y = fmaf(nj, v1.y, a1.y);
        a1.z = fmaf(nj, v1.z, a1.z); a1.w = fmaf(nj, v1.w, a1.w);
    }
    if (doRelu) {
        a0.x = fmaxf(a0.x, 0.f); a0.y = fmaxf(a0.y, 0.f);
        a0.z = fmaxf(a0.z, 0.f); a0.w = fmaxf(a0.w, 0.f);
        a1.x = fmaxf(a1.x, 0.f); a1.y = fmaxf(a1.y, 0.f);
        a1.z = fmaxf(a1.z, 0.f); a1.w = fmaxf(a1.w, 0.f);
    }
    float4* op = (float4*)(out + (size_t)node * 256);
    op[c0] = a0;
    op[c1] = a1;
}

// ------------------------- CSR gather aggregation, F=64, one wave per node
__global__ void gather64_kernel(const int* __restrict__ rowstart,
                                const int* __restrict__ csr_src,
                                const float* __restrict__ csr_norm,
                                const float* __restrict__ dinv,
                                const float* __restrict__ h,
                                const float* __restrict__ bias,
                                float* __restrict__ out, int n) {
    const int lane = threadIdx.x & 31;
    int node = (blockIdx.x * blockDim.x + threadIdx.x) >> 5;
    if (node >= n) return;
    node = __builtin_amdgcn_readfirstlane(node);

    const int beg = rowstart[node];
    const int end = rowstart[node + 1];
    const float di = dinv[node];
    const float s  = di * di;

    const float2* hn = (const float2*)(h + (size_t)node * 64);
    const float2* b2 = (const float2*)bias;
    float2 hv = hn[lane];
    float2 bv = b2[lane];
    float2 a;
    a.x = fmaf(s, hv.x, bv.x);
    a.y = fmaf(s, hv.y, bv.y);

    for (int j = beg; j < end; ++j) {
        const int   sj = csr_src[j];
        const float nj = csr_norm[j];
        const float2* hs = (const float2*)(h + (size_t)sj * 64);
        float2 v = hs[lane];
        a.x = fmaf(nj, v.x, a.x);
        a.y = fmaf(nj, v.y, a.y);
    }
    ((float2*)(out + (size_t)node * 64))[lane] = a;
}

// --------------------------------------------------------------------------
extern "C" void kernel_launch(void* const* d_in, const int* in_sizes, int n_in,
                              void* d_out, int out_size, void* d_ws, size_t ws_size,
                              hipStream_t stream) {
    (void)n_in; (void)out_size; (void)ws_size;

    const float* x  = (const float*)d_in[0];
    const int*   ei = (const int*)d_in[1];        // int32 edge_index [2, E]
    const float* W1 = (const float*)d_in[2];
    const float* b1 = (const float*)d_in[3];
    const float* W2 = (const float*)d_in[4];
    const float* b2 = (const float*)d_in[5];
    float* out = (float*)d_out;

    const int Fin = 128, Fmid = 256, Fout = 64;
    const int N = in_sizes[0] / Fin;              // 50000
    const int E = in_sizes[1] / 2;                // 800000
    const int* src = ei;
    const int* dst = ei + E;

    // workspace layout (256B aligned)
    char* ws = (char*)d_ws;
    size_t off = 0;
    auto take = [&](size_t bytes) -> char* {
        char* p = ws + off;
        off = (off + bytes + 255) & ~(size_t)255;
        return p;
    };
    int*   deg      = (int*)  take((size_t)N * 4);
    float* dinv     = (float*)take((size_t)N * 4);
    int*   rowstart = (int*)  take((size_t)(N + 1) * 4);
    int*   cursor   = (int*)  take((size_t)N * 4);
    int*   csr_src  = (int*)  take((size_t)E * 4);
    float* csr_norm = (float*)take((size_t)E * 4);
    float* hbuf     = (float*)take((size_t)N * Fmid * 4);   // h1, later reused as h2
    float* hrelu    = (float*)take((size_t)N * Fmid * 4);
    float* h2 = hbuf;

    const int T = 256;                                       // 8 waves / block

    // 1) degrees + dinv + CSR
    zero_i32_kernel <<<(N + T - 1) / T, T, 0, stream>>>(deg, N);
    count_deg_kernel<<<(E + T - 1) / T, T, 0, stream>>>(dst, deg, E);
    dinv_kernel     <<<(N + T - 1) / T, T, 0, stream>>>(deg, dinv, N);
    scan_kernel     <<<1, T, 0, stream>>>(deg, rowstart, N);
    copy_i32_kernel <<<(N + T - 1) / T, T, 0, stream>>>(rowstart, cursor, N);
    fill_csr_kernel <<<(E + T - 1) / T, T, 0, stream>>>(src, dst, dinv, cursor,
                                                        csr_src, csr_norm, E);

    // 2) h1 = x @ W1   (WMMA fp32)
    {
        int tiles = (N / 16) * (Fmid / 16);
        int wpb = T / 32;
        gemm_f32_wmma_kernel<<<(tiles + wpb - 1) / wpb, T, 0, stream>>>(x, W1, hbuf, N, Fmid, Fin);
    }
    // 3) hrelu = relu(b1 + dinv^2*h1 + CSR-gather(norm*h1))   -- fused
    {
        long long threads = (long long)N * 32;
        gather256_kernel<<<(int)((threads + T - 1) / T), T, 0, stream>>>(
            rowstart, csr_src, csr_norm, dinv, hbuf, b1, hrelu, N, 1);
    }
    // 4) h2 = hrelu @ W2   (WMMA fp32)
    {
        int tiles = (N / 16) * (Fout / 16);
        int wpb = T / 32;
        gemm_f32_wmma_kernel<<<(tiles + wpb - 1) / wpb, T, 0, stream>>>(hrelu, W2, h2, N, Fout, Fmid);
    }
    // 5) out = b2 + dinv^2*h2 + CSR-gather(norm*h2)           -- fused
    {
        long long threads = (long long)N * 32;
        gather64_kernel<<<(int)((threads + T - 1) / T), T, 0, stream>>>(
            rowstart, csr_src, csr_norm, dinv, h2, b2, out, N);
    }
}